// GCN_6476810682617
// MI455X (gfx1250) — compile-verified
//
#include <hip/hip_runtime.h>
#include <math.h>

#define N_NODES 100000
#define N_EDGES 3200000
#define F_IN    512
#define HID     16
#define NTILES  (N_NODES / 16)   // 6250, exact

typedef __attribute__((ext_vector_type(2))) float v2f;
typedef __attribute__((ext_vector_type(8))) float v8f;

// no-return hardware fp32 atomic add (global_atomic_add_f32)
__device__ __forceinline__ void atomic_fadd(float* p, float v) {
    __hip_atomic_fetch_add(p, v, __ATOMIC_RELAXED, __HIP_MEMORY_SCOPE_AGENT);
}

// ---------------- degree / normalization (integer atomics) ----------------

__global__ void init_cnt_kernel(int* __restrict__ cnt, int n) {
    int i = blockIdx.x * blockDim.x + threadIdx.x;
    if (i < n) cnt[i] = 1;   // self-loop
}

__global__ void accum_cnt_kernel(const int* __restrict__ dst, int* __restrict__ cnt, int e) {
    int i = blockIdx.x * blockDim.x + threadIdx.x;
    if (i < e) atomicAdd(&cnt[__builtin_nontemporal_load(dst + i)], 1);
}

// convert in place: int count -> float rsqrt(count)
__global__ void dinv_kernel(int* __restrict__ cnt, float* __restrict__ dinv, int n) {
    int i = blockIdx.x * blockDim.x + threadIdx.x;
    if (i < n) {
        float d = (float)cnt[i];
        dinv[i] = rsqrtf(d);   // count >= 1 always (self-loop)
    }
}

// ---------------- GEMM1: h = x @ W1 via fp32 WMMA, fused self-loop init ----------------
// One wave per 16-row tile; 8 waves (256 threads) per block.
// W1 (512x16 fp32 = 32KB) staged in LDS per block. x streamed with NT hint.

__global__ __launch_bounds__(256) void gemm1_wmma_kernel(
        const float* __restrict__ x, const float* __restrict__ W1,
        const float* __restrict__ dinv,
        float* __restrict__ h, float* __restrict__ acc1) {
    __shared__ float ldsW[F_IN * HID];
    for (int i = threadIdx.x; i < F_IN * HID; i += 256) ldsW[i] = W1[i];
    __syncthreads();

    const int lane = threadIdx.x & 31;
    const int wave = threadIdx.x >> 5;
    const int tile = blockIdx.x * 8 + wave;
    if (tile >= NTILES) return;          // wave-uniform: EXEC all-1s inside

    const int row0 = tile * 16;
    const int mcol = lane & 15;          // M for A, N-column for B
    const int kk   = (lane >> 4) << 1;   // lanes 0-15: K=0,1 ; lanes 16-31: K=2,3

    const float* __restrict__ xrow = x + (size_t)(row0 + mcol) * F_IN + kk;   // 8B aligned
    const float* __restrict__ wptr = ldsW + kk * HID + mcol;

    v8f c = {0.f, 0.f, 0.f, 0.f, 0.f, 0.f, 0.f, 0.f};
#pragma unroll 8
    for (int k0 = 0; k0 < F_IN; k0 += 4) {
        v2f a = __builtin_nontemporal_load((const v2f*)(xrow + k0));  // global_load_b64 NT
        v2f b;
        b.x = wptr[k0 * HID];                      // ds loads, conflict-free
        b.y = wptr[k0 * HID + HID];
        c = __builtin_amdgcn_wmma_f32_16x16x4_f32(
                /*neg_a=*/false, a, /*neg_b=*/false, b,
                /*c_mod=*/(short)0, c, /*reuse_a=*/false, /*reuse_b=*/false);
    }

    // C/D layout: VGPR v -> row 8*(lane>=16)+v, col lane&15
    // Fused epilogue: write h and the layer-1 self-loop term acc1 = h * dinv^2.
    const int rb = row0 + (lane >> 4) * 8;
    float* __restrict__ hp = h    + (size_t)rb * HID + mcol;
    float* __restrict__ ap = acc1 + (size_t)rb * HID + mcol;
#pragma unroll
    for (int v = 0; v < 8; v++) {
        float hv = c[v];
        float di = dinv[rb + v];
        hp[v * HID] = hv;
        ap[v * HID] = hv * di * di;
    }
}

// ---------------- layer-1 edge aggregation ----------------
// One thread per (edge, feature): each wave = 2 edges x 16 contiguous features.
// Gather is a coalesced b32 load; scatter is ONE atomic instruction per lane,
// with the 16 lanes of an edge hitting 64 contiguous bytes.

__global__ __launch_bounds__(256) void edge_agg1_kernel(
        const int* __restrict__ src, const int* __restrict__ dst,
        const float* __restrict__ dinv, const float* __restrict__ h,
        float* __restrict__ acc1, int total /* = E*16 */) {
    int i = blockIdx.x * blockDim.x + threadIdx.x;
    if (i >= total) return;
    int e = i >> 4;
    int f = i & 15;
    int s = src[e];            // broadcast within wave (16 lanes share)
    int d = dst[e];
    float w = dinv[s] * dinv[d];
    float hv = h[(size_t)s * HID + f];
    atomic_fadd(acc1 + (size_t)d * HID + f, hv * w);
}

// ---------------- per-node: relu+bias, collapsed layer-2 GEMM (16->1), self-loop init ----------------

__global__ void layer2_node_kernel(const float* __restrict__ acc1, const float* __restrict__ b1,
                                   const float* __restrict__ W2, const float* __restrict__ dinv,
                                   float* __restrict__ z, float* __restrict__ acc2, int n) {
    int i = blockIdx.x * blockDim.x + threadIdx.x;
    if (i >= n) return;
    float zz = 0.0f;
    const float* __restrict__ row = acc1 + (size_t)i * HID;
#pragma unroll
    for (int f = 0; f < HID; f++) {
        float r = fmaxf(row[f] + b1[f], 0.0f);     // ReLU(agg1 + b1)
        zz += r * W2[f];                           // (h1 @ W2) scalar
    }
    z[i] = zz;
    float di = dinv[i];
    acc2[i] = zz * di * di;                        // self-loop term of layer 2
}

__global__ void edge_agg2_kernel(const int* __restrict__ src, const int* __restrict__ dst,
                                 const float* __restrict__ dinv, const float* __restrict__ z,
                                 float* __restrict__ acc2, int e) {
    int i = blockIdx.x * blockDim.x + threadIdx.x;
    if (i >= e) return;
    int s = __builtin_nontemporal_load(src + i);
    int d = __builtin_nontemporal_load(dst + i);
    atomic_fadd(&acc2[d], z[s] * dinv[s] * dinv[d]);
}

__global__ void sigmoid_kernel(const float* __restrict__ acc2, const float* __restrict__ b2,
                               float* __restrict__ out, int n) {
    int i = blockIdx.x * blockDim.x + threadIdx.x;
    if (i < n) {
        float v = acc2[i] + b2[0];
        out[i] = 1.0f / (1.0f + expf(-v));
    }
}

// ---------------- launcher ----------------

extern "C" void kernel_launch(void* const* d_in, const int* in_sizes, int n_in,
                              void* d_out, int out_size, void* d_ws, size_t ws_size,
                              hipStream_t stream) {
    const float* x  = (const float*)d_in[0];
    const float* W1 = (const float*)d_in[1];
    const float* b1 = (const float*)d_in[2];
    const float* W2 = (const float*)d_in[3];
    const float* b2 = (const float*)d_in[4];
    const int*   ei = (const int*)d_in[5];
    const int* src = ei;
    const int* dst = ei + N_EDGES;
    float* out = (float*)d_out;

    // workspace layout (floats): dinv[N] | h[N*16] | acc1[N*16] | z[N] | acc2[N]  (~14MB)
    float* ws   = (float*)d_ws;
    float* dinv = ws;
    int*   cnt  = (int*)ws;                 // aliases dinv; converted in place
    float* h    = dinv + N_NODES;
    float* acc1 = h    + (size_t)N_NODES * HID;
    float* z    = acc1 + (size_t)N_NODES * HID;
    float* acc2 = z    + N_NODES;

    const int T = 256;
    const int gN   = (N_NODES + T - 1) / T;
    const int gE   = (N_EDGES + T - 1) / T;
    const int gEF  = (N_EDGES * HID + T - 1) / T;   // 51.2M threads, fits int
    const int gTile = (NTILES + 7) / 8;

    init_cnt_kernel  <<<gN,   T, 0, stream>>>(cnt, N_NODES);
    accum_cnt_kernel <<<gE,   T, 0, stream>>>(dst, cnt, N_EDGES);
    dinv_kernel      <<<gN,   T, 0, stream>>>(cnt, dinv, N_NODES);

    gemm1_wmma_kernel<<<gTile, T, 0, stream>>>(x, W1, dinv, h, acc1);

    edge_agg1_kernel <<<gEF,  T, 0, stream>>>(src, dst, dinv, h, acc1, N_EDGES * HID);

    layer2_node_kernel<<<gN,  T, 0, stream>>>(acc1, b1, W2, dinv, z, acc2, N_NODES);
    edge_agg2_kernel <<<gE,   T, 0, stream>>>(src, dst, dinv, z, acc2, N_EDGES);

    sigmoid_kernel   <<<gN,   T, 0, stream>>>(acc2, b2, out, N_NODES);
}